// ElectrostaticsModule_11699490915122
// MI455X (gfx1250) — compile-verified
//
#include <hip/hip_runtime.h>
#include <hip/hip_bf16.h>

typedef __attribute__((ext_vector_type(2))) float v2f;
typedef __attribute__((ext_vector_type(8))) float v8f;

// Problem constants (fixed by the reference's setup_inputs)
constexpr int   Bb    = 4;
constexpr int   Nn    = 2048;
constexpr int   Cc    = 32;
constexpr int   NT    = Nn / 16;      // 128 tiles per dimension
constexpr int   STRIP = 8;            // j-tiles handled per wave
constexpr int   NJS   = NT / STRIP;   // 16 j-strips
constexpr float CONVF = 332.07156f;

// ---------------------------------------------------------------------------
// Kernel 1: fold embs @ w_j and embs @ w_i to per-(b,n) scalars; zero acc.
// ---------------------------------------------------------------------------
__global__ void es_pre(const float* __restrict__ embs,
                       const float* __restrict__ sfw,
                       float* __restrict__ aI, float* __restrict__ aJ,
                       float* __restrict__ acc)
{
    int t = blockIdx.x * blockDim.x + threadIdx.x;
    if (t == 0) acc[0] = 0.0f;
    if (t < Bb * Nn) {
        const float* e = embs + (size_t)t * Cc;
        float sj = 0.0f, si = 0.0f;
        #pragma unroll
        for (int c = 0; c < Cc; ++c) {
            float ev = e[c];
            sj = fmaf(ev, sfw[c],      sj);   // w_j = w[0:C]
            si = fmaf(ev, sfw[Cc + c], si);   // w_i = w[C:2C]
        }
        aJ[t] = sj;
        aI[t] = si;
    }
}

// ---------------------------------------------------------------------------
// Kernel 2: main pairwise energy. One wave per (b, i-tile, j-strip).
// Pair->lane mapping == WMMA f32 16x16 C/D layout:
//   VGPR v, lane L : i_local = v + 8*(L>=16), j_local = L & 15
// OPP (Ps_i . Ps_j) comes straight out of v_wmma_f32_16x16x4_f32 in that
// same layout, so it is consumed in-register.
// ---------------------------------------------------------------------------
__global__ __launch_bounds__(32) void es_main(
    const float* __restrict__ X,   const float* __restrict__ qs,
    const float* __restrict__ Ps,  const float* __restrict__ pm,
    const float* __restrict__ sfw,
    const float* __restrict__ aI,  const float* __restrict__ aJ,
    float* __restrict__ acc)
{
    const int lane = threadIdx.x;
    const int l15  = lane & 15;
    const int hi   = lane >> 4;          // 0 for lanes 0-15, 1 for 16-31

    int bid = blockIdx.x;
    const int js = bid % NJS;  bid /= NJS;
    const int it = bid % NT;   bid /= NT;
    const int b  = bid;
    const int i0 = it * 16;

    const float wd = sfw[2 * Cc];        // w_d = w[-1]

    // Stage i-side data once per wave (512 B of LDS).
    __shared__ float4 sI1[16];           // (Xi.x, Xi.y, Xi.z, qs_i)
    __shared__ float4 sI2[16];           // (Pi.x, Pi.y, Pi.z, a_i)
    if (lane < 16) {
        const int i = i0 + lane;
        const float* xp = X + ((size_t)b * Nn + i) * 3;
        sI1[lane] = make_float4(xp[0], xp[1], xp[2], qs[i]);
        const float* pp = Ps + (size_t)i * 3;
        sI2[lane] = make_float4(pp[0], pp[1], pp[2], aI[(size_t)b * Nn + i]);
    }
    __syncthreads();

    // WMMA A operand: Ps_i as 16x4 f32 (K=3 padded with 0).
    // ISA layout: lanes 0-15 hold K0/K1, lanes 16-31 hold K2/K3.
    // A's K3 column is hard zero -> B's K3 row may hold garbage.
    const float4 pA = sI2[l15];
    v2f amat;
    amat.x = hi ? pA.z : pA.x;
    amat.y = hi ? 0.0f : pA.y;

    float lsum = 0.0f;

    const int jt_end = js * STRIP + STRIP;
    for (int jt = js * STRIP; jt < jt_end; ++jt) {
        const int j = jt * 16 + l15;

        // Per-lane j-side data (all loads unconditional, no divergence).
        const float* xjp = X + ((size_t)b * Nn + j) * 3;
        const float xjx = xjp[0], xjy = xjp[1], xjz = xjp[2];
        const float qj  = qs[j];
        const float ajv = aJ[(size_t)b * Nn + j];
        const float* pjp = Ps + (size_t)j * 3;
        const float pjx = pjp[0], pjy = pjp[1], pjz = pjp[2];

        // WMMA B operand: Ps_j as 4x16 f32. K3 row (hi-lane .y) pairs with
        // A's zeroed K3 column, so pjy there is harmless -> no select needed.
        v2f bmat;
        bmat.x = hi ? pjz : pjx;
        bmat.y = pjy;

        // OPP tile = Ps_i (16x4) x Ps_j^T (4x16), f32 WMMA, C = 0.
        v8f c0 = {};
        v8f opp = __builtin_amdgcn_wmma_f32_16x16x4_f32(
            false, amat, false, bmat, (short)0, c0, false, false);

        // Mask tile base: row (i0 + 8*hi + v), col j.
        const float* pmrow =
            pm + ((size_t)b * Nn + (size_t)(i0 + 8 * hi)) * Nn + j;

        if (jt + 1 < jt_end) {
            #pragma unroll
            for (int v = 0; v < 8; ++v)
                __builtin_prefetch(pmrow + (size_t)v * Nn + 16, 0, 1);
        }

        #pragma unroll
        for (int v = 0; v < 8; ++v) {
            const float4 I1 = sI1[v + 8 * hi];   // Xi, qi  (LDS broadcast)
            const float4 I2 = sI2[v + 8 * hi];   // Pi, ai

            // V = X[j] - X[i]
            const float dx = xjx - I1.x;
            const float dy = xjy - I1.y;
            const float dz = xjz - I1.z;

            // D = sqrt(sum(V*V + 1e-6)) = sqrt(|V|^2 + 3e-6)
            const float r2   = fmaf(dx, dx, fmaf(dy, dy, fmaf(dz, dz, 3e-6f)));
            const float Dd   = sqrtf(r2);
            const float invD = 1.0f / (Dd + 1e-6f);

            // Vn = (V + 1e-6) / max(||V + 1e-6||, 1e-12)
            const float vx = dx + 1e-6f, vy = dy + 1e-6f, vz = dz + 1e-6f;
            const float n2 = fmaf(vx, vx, fmaf(vy, vy, vz * vz));
            const float rn = 1.0f / fmaxf(sqrtf(n2), 1e-12f);

            // OqP = Vn . Ps[i]
            const float oqp = fmaf(vx, I2.x, fmaf(vy, I2.y, vz * I2.z)) * rn;

            // sf = a_j[b,j] + a_i[b,i] + invD * w_d
            const float sf = ajv + I2.w + invD * wd;

            const float invD2 = invD * invD;
            const float invD4 = invD2 * invD2;
            const float invD6 = invD4 * invD2;

            // E = qi*qj*invD*sf + OqP*invD^4 + OPP*invD^6
            const float e = fmaf(I1.w * qj * invD, sf,
                             fmaf(oqp, invD4, opp[v] * invD6));

            lsum = fmaf(pmrow[(size_t)v * Nn], e, lsum);
        }
    }

    // Wave32 XOR-butterfly reduction via ds_swizzle (group-of-32 mode:
    // offset = xor_mask<<10 | 0x1f). No lane clamps, no bpermute.
    lsum += __int_as_float(__builtin_amdgcn_ds_swizzle(__float_as_int(lsum), 0x401f)); // xor 16
    lsum += __int_as_float(__builtin_amdgcn_ds_swizzle(__float_as_int(lsum), 0x201f)); // xor 8
    lsum += __int_as_float(__builtin_amdgcn_ds_swizzle(__float_as_int(lsum), 0x101f)); // xor 4
    lsum += __int_as_float(__builtin_amdgcn_ds_swizzle(__float_as_int(lsum), 0x081f)); // xor 2
    lsum += __int_as_float(__builtin_amdgcn_ds_swizzle(__float_as_int(lsum), 0x041f)); // xor 1
    if (lane == 0)
        atomicAdd(acc, lsum);
}

// ---------------------------------------------------------------------------
// Kernel 3: scale + NaN guard.
// ---------------------------------------------------------------------------
__global__ void es_finalize(const float* __restrict__ acc,
                            float* __restrict__ out)
{
    const float t = acc[0] * (0.5f * CONVF);
    out[0] = __builtin_isnan(t) ? 1e-6f : t;
}

// ---------------------------------------------------------------------------
extern "C" void kernel_launch(void* const* d_in, const int* in_sizes, int n_in,
                              void* d_out, int out_size, void* d_ws, size_t ws_size,
                              hipStream_t stream)
{
    (void)in_sizes; (void)n_in; (void)out_size; (void)ws_size;

    const float* X    = (const float*)d_in[0];   // (B,N,3)
    const float* embs = (const float*)d_in[1];   // (B,N,C)
    const float* qs   = (const float*)d_in[2];   // (N,)
    const float* Ps   = (const float*)d_in[3];   // (N,3)
    const float* pm   = (const float*)d_in[4];   // (B,N,N)
    const float* sfw  = (const float*)d_in[5];   // (2C+1,1)
    float* out = (float*)d_out;

    // Workspace layout: [0] accumulator, [16..) a_i (B*N), then a_j (B*N).
    float* acc = (float*)d_ws;
    float* aI  = (float*)d_ws + 16;
    float* aJ  = aI + Bb * Nn;

    es_pre<<<(Bb * Nn + 255) / 256, 256, 0, stream>>>(embs, sfw, aI, aJ, acc);
    es_main<<<Bb * NT * NJS, 32, 0, stream>>>(X, qs, Ps, pm, sfw, aI, aJ, acc);
    es_finalize<<<1, 1, 0, stream>>>(acc, out);
}